// GroupAwareContrastiveLoss_68023692034360
// MI455X (gfx1250) — compile-verified
//
#include <hip/hip_runtime.h>

typedef float v2f __attribute__((ext_vector_type(2)));
typedef float v8f __attribute__((ext_vector_type(8)));
typedef int   vi4 __attribute__((vector_size(4 * sizeof(int))));   // 16B vector, matches builtin param

#define TILE      128      // block tile (M and N)
#define BK        16       // K slab per LDS stage (double-buffered)
#define LDW       20       // padded LDS row stride in floats (multiple of 4 -> 16B-aligned float4)
#define NTHREADS  256      // 8 waves (wave32)

static constexpr float K_M_POS = 0.5f;
static constexpr float K_M_NEG = 0.1f;

#define AS1 __attribute__((address_space(1)))
#define AS3 __attribute__((address_space(3)))

#if __has_builtin(__builtin_amdgcn_global_load_async_to_lds_b128)
#define HAVE_ASYNC_LDS 1
#else
#define HAVE_ASYNC_LDS 0
#warning "CDNA5: __builtin_amdgcn_global_load_async_to_lds_b128 not available -> sync float4 fallback"
#endif

__device__ __forceinline__ int imin(int a, int b) { return a < b ? a : b; }

// One 16-byte global -> LDS transfer per lane.
__device__ __forceinline__ void copy_b128(const float* __restrict__ g, float* l) {
#if HAVE_ASYNC_LDS
    __builtin_amdgcn_global_load_async_to_lds_b128(
        (AS1 vi4*)(void*)g, (AS3 vi4*)(void*)l, 0, 0);
#else
    *(float4*)l = *(const float4*)g;
    __builtin_prefetch(g + BK, 0, 3);
#endif
}

__device__ __forceinline__ void wait_async_all() {
#if HAVE_ASYNC_LDS
#if __has_builtin(__builtin_amdgcn_s_wait_asynccnt)
    __builtin_amdgcn_s_wait_asynccnt(0);
#else
    asm volatile("s_wait_asynccnt 0x0" ::: "memory");
#endif
#endif
}

// ---------------------------------------------------------------------------
// Kernel 1: per-row squared norm + reciprocal norm. One block per row.
// ---------------------------------------------------------------------------
__global__ __launch_bounds__(256)
void gacl_row_stats(const float* __restrict__ cb, float* __restrict__ sq,
                    float* __restrict__ rn, int N, int D) {
    __shared__ float red[256];
    const int row = blockIdx.x;
    const float* p = cb + (size_t)row * D;
    float s = 0.f;
    for (int c = threadIdx.x; c < D; c += 256) {
        float x = p[c];
        s += x * x;
    }
    red[threadIdx.x] = s;
    __syncthreads();
    for (int st = 128; st > 0; st >>= 1) {
        if ((int)threadIdx.x < st) red[threadIdx.x] += red[threadIdx.x + st];
        __syncthreads();
    }
    if (threadIdx.x == 0) {
        float t = red[0];
        sq[row] = t;
        rn[row] = rsqrtf(t);
    }
}

// ---------------------------------------------------------------------------
// Kernel 2: count valid rows (pos_cnt>0 && neg_cnt>0, i < Mrows). One block.
// ---------------------------------------------------------------------------
__global__ __launch_bounds__(1024)
void gacl_valid_count(const int* __restrict__ starts, const int* __restrict__ ends,
                      const int* __restrict__ maxi, int N, int* __restrict__ outCnt) {
    __shared__ int red[1024];
    const int Mrows = imin(N, maxi[0] + 1);
    int cnt = 0;
    for (int i = threadIdx.x; i < Mrows; i += 1024) {
        int st = starts[i], en = ends[i];
        int cin = en - st + 1; if (cin < 0) cin = 0;
        int selfIn = (i >= st && i <= en) ? 1 : 0;
        int pcnt = cin - selfIn;
        int ncnt = N - cin + selfIn;
        if (pcnt > 0 && ncnt > 0) cnt++;
    }
    red[threadIdx.x] = cnt;
    __syncthreads();
    for (int st = 512; st > 0; st >>= 1) {
        if ((int)threadIdx.x < st) red[threadIdx.x] += red[threadIdx.x + st];
        __syncthreads();
    }
    if (threadIdx.x == 0) *outCnt = red[0];
}

// ---------------------------------------------------------------------------
// Kernel 3: double-buffered async-copy WMMA fp32 Gram GEMM + fused epilogue.
// Grid: (N/TILE)^2 blocks, each emits one deterministic partial sum.
// ---------------------------------------------------------------------------
__global__ __launch_bounds__(NTHREADS)
void gacl_gram_loss_tile(const float* __restrict__ cb,
                         const int* __restrict__ starts,
                         const int* __restrict__ ends,
                         const int* __restrict__ maxi,
                         const float* __restrict__ sq,
                         const float* __restrict__ rn,
                         float* __restrict__ partials,
                         int N, int D, int tilesPerSide) {
    __shared__ float As[2][TILE * LDW];
    __shared__ float Bs[2][TILE * LDW];
    __shared__ float sRow[TILE], rRow[TILE], sCol[TILE], rCol[TILE];
    __shared__ int   stR[TILE], enR[TILE];
    __shared__ float red[NTHREADS];

    const int bx = blockIdx.x % tilesPerSide;   // column tile
    const int by = blockIdx.x / tilesPerSide;   // row tile
    const int rowBase = by * TILE;
    const int colBase = bx * TILE;
    const int tid   = threadIdx.x;
    const int lane  = tid & 31;
    const int wave  = tid >> 5;
    const int waveM = wave >> 2;                // 0..1  (64 rows each)
    const int waveN = wave & 3;                 // 0..3  (32 cols each)
    const int lr    = lane & 15;
    const int hi    = lane >> 4;                // lane group: K offset +2 / M offset +8
    const int Mrows = imin(N, maxi[0] + 1);

    // Per-thread copy coordinates: 128 rows x 16 floats = 512 float4 per matrix,
    // 2 transfers per thread per matrix per slab.
    const int cpR0 = tid >> 2;                  // rows 0..63
    const int cpR1 = cpR0 + 64;                 // rows 64..127
    const int cpC  = (tid & 3) << 2;            // 0,4,8,12

    // Stage per-row / per-col stats for the fused epilogue.
    for (int t = tid; t < TILE; t += NTHREADS) {
        int ri = rowBase + t, ci = colBase + t;
        sRow[t] = sq[ri];  rRow[t] = rn[ri];
        stR[t]  = starts[ri]; enR[t] = ends[ri];
        sCol[t] = sq[ci];  rCol[t] = rn[ci];
    }

    v8f acc[4][2] = {};   // 64x32 per wave as 4x2 grid of 16x16 fp32 tiles

    const int nslab = D / BK;

    // Prologue: issue slab 0 into buffer 0.
    {
        const float* aBase = cb + (size_t)rowBase * D;
        const float* bBase = cb + (size_t)colBase * D;
        copy_b128(aBase + (size_t)cpR0 * D + cpC, &As[0][cpR0 * LDW + cpC]);
        copy_b128(aBase + (size_t)cpR1 * D + cpC, &As[0][cpR1 * LDW + cpC]);
        copy_b128(bBase + (size_t)cpR0 * D + cpC, &Bs[0][cpR0 * LDW + cpC]);
        copy_b128(bBase + (size_t)cpR1 * D + cpC, &Bs[0][cpR1 * LDW + cpC]);
    }

    for (int s = 0; s < nslab; ++s) {
        const int buf = s & 1;
        wait_async_all();        // this wave's slab-s transfers have landed in LDS
        __syncthreads();         // all waves' transfers visible; buf^1 consumers done

        if (s + 1 < nslab) {     // overlap: slab s+1 flies while we run WMMAs on slab s
            const int k0n = (s + 1) * BK;
            const float* aBase = cb + (size_t)rowBase * D + k0n;
            const float* bBase = cb + (size_t)colBase * D + k0n;
            const int nb = buf ^ 1;
            copy_b128(aBase + (size_t)cpR0 * D + cpC, &As[nb][cpR0 * LDW + cpC]);
            copy_b128(aBase + (size_t)cpR1 * D + cpC, &As[nb][cpR1 * LDW + cpC]);
            copy_b128(bBase + (size_t)cpR0 * D + cpC, &Bs[nb][cpR0 * LDW + cpC]);
            copy_b128(bBase + (size_t)cpR1 * D + cpC, &Bs[nb][cpR1 * LDW + cpC]);
        }

        // 4 WMMA K-substeps of 4 over the slab.
#pragma unroll
        for (int kk = 0; kk < BK; kk += 4) {
            const int kq = kk + 2 * hi;         // fp32 A/B frag: v0=K{0,2}, v1=K{1,3}
            v2f bfrag[2];
#pragma unroll
            for (int nt = 0; nt < 2; ++nt) {
                int col = waveN * 32 + nt * 16 + lr;
                bfrag[nt] = *(const v2f*)&Bs[buf][col * LDW + kq];
            }
#pragma unroll
            for (int mt = 0; mt < 4; ++mt) {
                int row = waveM * 64 + mt * 16 + lr;
                v2f afrag = *(const v2f*)&As[buf][row * LDW + kq];
#pragma unroll
                for (int nt = 0; nt < 2; ++nt) {
                    acc[mt][nt] = __builtin_amdgcn_wmma_f32_16x16x4_f32(
                        false, afrag, false, bfrag[nt],
                        (short)0, acc[mt][nt], false, false);
                }
            }
        }
    }

    // Fused epilogue: masked pos/neg loss terms, pre-divided by per-row counts.
    float partial = 0.f;
#pragma unroll
    for (int mt = 0; mt < 4; ++mt) {
#pragma unroll
        for (int nt = 0; nt < 2; ++nt) {
            const int jLoc = waveN * 32 + nt * 16 + lr;
            const int j    = colBase + jLoc;
            const float sqj = sCol[jLoc];
            const float rnj = rCol[jLoc];
            v8f c = acc[mt][nt];
#pragma unroll
            for (int v = 0; v < 8; ++v) {
                const int iLoc = waveM * 64 + mt * 16 + v + 8 * hi;  // C/D layout: M=v (+8 for hi lanes)
                const int i    = rowBase + iLoc;
                if (i >= Mrows) continue;
                const float dot = c[v];
                const int st = stR[iLoc], en = enR[iLoc];
                int cin = en - st + 1; if (cin < 0) cin = 0;
                const int selfIn = (i >= st && i <= en) ? 1 : 0;
                const int pcnt = cin - selfIn;
                const int ncnt = N - cin + selfIn;
                if (!(pcnt > 0 && ncnt > 0)) continue;               // row invalid -> 0
                const bool inr  = (j >= st) && (j <= en);
                const bool self = (j == i);
                if (inr && !self) {
                    float d2 = fmaxf(sRow[iLoc] + sqj - 2.0f * dot, 0.0f);
                    float t  = sqrtf(d2) - K_M_POS;
                    if (t > 0.0f) partial += (t * t) / (float)pcnt;
                } else {
                    float cosv = dot * rRow[iLoc] * rnj;
                    cosv = fminf(fmaxf(cosv, -1.0f), 1.0f);
                    float t = fabsf(cosv) - K_M_NEG;
                    if (t > 0.0f) partial += (t * t) / (float)ncnt;
                }
            }
        }
    }

    red[tid] = partial;
    __syncthreads();
    for (int s = NTHREADS / 2; s > 0; s >>= 1) {
        if (tid < s) red[tid] += red[tid + s];
        __syncthreads();
    }
    if (tid == 0) partials[blockIdx.x] = red[0];
}

// ---------------------------------------------------------------------------
// Kernel 4: deterministic final reduction of block partials -> scalar loss.
// ---------------------------------------------------------------------------
__global__ __launch_bounds__(256)
void gacl_finalize(const float* __restrict__ partials, int nblk,
                   const int* __restrict__ validCnt, float* __restrict__ out) {
    __shared__ float red[256];
    float s = 0.f;
    for (int t = threadIdx.x; t < nblk; t += 256) s += partials[t];
    red[threadIdx.x] = s;
    __syncthreads();
    for (int st = 128; st > 0; st >>= 1) {
        if ((int)threadIdx.x < st) red[threadIdx.x] += red[threadIdx.x + st];
        __syncthreads();
    }
    if (threadIdx.x == 0) {
        int c = *validCnt;
        out[0] = (c > 0) ? red[0] / (float)c : 0.0f;
    }
}

// ---------------------------------------------------------------------------
extern "C" void kernel_launch(void* const* d_in, const int* in_sizes, int n_in,
                              void* d_out, int out_size, void* d_ws, size_t ws_size,
                              hipStream_t stream) {
    const float* cb     = (const float*)d_in[0];
    const int*   starts = (const int*)d_in[1];
    const int*   ends   = (const int*)d_in[2];
    const int*   maxi   = (const int*)d_in[3];

    const int N = in_sizes[1];            // 8192
    const int D = in_sizes[0] / N;        // 1024
    const int tps  = N / TILE;            // 64
    const int nblk = tps * tps;           // 4096

    float* sq       = (float*)d_ws;
    float* rn       = sq + N;
    float* partials = rn + N;
    int*   validCnt = (int*)(partials + nblk);

    gacl_row_stats<<<N, 256, 0, stream>>>(cb, sq, rn, N, D);
    gacl_valid_count<<<1, 1024, 0, stream>>>(starts, ends, maxi, N, validCnt);
    gacl_gram_loss_tile<<<nblk, NTHREADS, 0, stream>>>(cb, starts, ends, maxi,
                                                       sq, rn, partials, N, D, tps);
    gacl_finalize<<<1, 256, 0, stream>>>(partials, nblk, validCnt, (float*)d_out);
}